// VEG_60644938219818
// MI455X (gfx1250) — compile-verified
//
#include <hip/hip_runtime.h>
#include <math.h>

// MI455X / gfx1250, wave32. Exp-bound P*G Gaussian field evaluation.
// Matmul part runs on the matrix pipe via V_WMMA_F32_16X16X4_F32 (fp32, the
// numerically correct choice since results feed exp); VALU does v_exp_f32.

typedef __attribute__((ext_vector_type(2))) float v2f;
typedef __attribute__((ext_vector_type(8))) float v8f;

#define P_N 65536
#define G_N 2048
#define LOG2E 1.44269504088896340736f

// ---------------------------------------------------------------------------
// Stage 1: per-Gaussian coefficient vector c[12], packed as float2 pairs in
// k-major layout cpack[pair][g] so stage-2 lanes read contiguous 128B runs.
//   power[p,g] = f[p] . c[g],  f = [x²,y²,z²,xy,xz,yz,x,y,z,1,0,0]
//   c = LOG2E * [-.5A00,-.5A11,-.5A22,-A01,-A02,-A12, Amu, -.5*const+log(wv), 0,0]
// ---------------------------------------------------------------------------
__global__ __launch_bounds__(256) void gs_precompute(
    const float* __restrict__ xyz, const float* __restrict__ weight,
    const float* __restrict__ scaling, const float* __restrict__ rotation,
    const float* __restrict__ values, float2* __restrict__ cpack)
{
    int g = blockIdx.x * blockDim.x + threadIdx.x;
    if (g >= G_N) return;

    // capped scales
    float sx = expf(scaling[g*3+0]);
    float sy = expf(scaling[g*3+1]);
    float sz = expf(scaling[g*3+2]);
    float r  = sqrtf(sx*sx + sy*sy + sz*sz) + 1e-8f;
    float rs = 0.02f * tanhf(r * (1.0f/0.02f));
    float k  = rs / r;
    sx *= k; sy *= k; sz *= k;
    float is0 = 1.0f/(sx*sx), is1 = 1.0f/(sy*sy), is2 = 1.0f/(sz*sz);

    // quaternion -> rotation matrix
    float qw = rotation[g*4+0], qx = rotation[g*4+1];
    float qy = rotation[g*4+2], qz = rotation[g*4+3];
    float qn = sqrtf(qw*qw + qx*qx + qy*qy + qz*qz) + 1e-12f;
    float inq = 1.0f/qn;
    qw *= inq; qx *= inq; qy *= inq; qz *= inq;
    float R00 = 1.f-2.f*(qy*qy+qz*qz), R01 = 2.f*(qx*qy-qw*qz), R02 = 2.f*(qx*qz+qw*qy);
    float R10 = 2.f*(qx*qy+qw*qz), R11 = 1.f-2.f*(qx*qx+qz*qz), R12 = 2.f*(qy*qz-qw*qx);
    float R20 = 2.f*(qx*qz-qw*qy), R21 = 2.f*(qy*qz+qw*qx), R22 = 1.f-2.f*(qx*qx+qy*qy);

    // A = R diag(1/s^2) R^T  (symmetric)
    float A00 = R00*R00*is0 + R01*R01*is1 + R02*R02*is2;
    float A01 = R00*R10*is0 + R01*R11*is1 + R02*R12*is2;
    float A02 = R00*R20*is0 + R01*R21*is1 + R02*R22*is2;
    float A11 = R10*R10*is0 + R11*R11*is1 + R12*R12*is2;
    float A12 = R10*R20*is0 + R11*R21*is1 + R12*R22*is2;
    float A22 = R20*R20*is0 + R21*R21*is1 + R22*R22*is2;

    float mx = xyz[g*3+0], my = xyz[g*3+1], mz = xyz[g*3+2];
    float Am0 = A00*mx + A01*my + A02*mz;
    float Am1 = A01*mx + A11*my + A12*mz;
    float Am2 = A02*mx + A12*my + A22*mz;
    float cst = mx*Am0 + my*Am1 + mz*Am2;

    // log(sigmoid(w)) + log(sigmoid(v)) folded into the exponent
    float lw = -log1pf(expf(-weight[g]));
    float lv = -log1pf(expf(-values[g]));

    float c[12];
    c[0] = -0.5f*A00; c[1] = -0.5f*A11; c[2] = -0.5f*A22;
    c[3] = -A01;      c[4] = -A02;      c[5] = -A12;
    c[6] = Am0;       c[7] = Am1;       c[8] = Am2;
    c[9] = -0.5f*cst + lw + lv;
    c[10] = 0.0f;     c[11] = 0.0f;

    // pre-scale by log2(e): epilogue is a bare v_exp_f32 (base-2 on AMD)
#pragma unroll
    for (int p = 0; p < 6; ++p)
        cpack[p*G_N + g] = make_float2(c[2*p]*LOG2E, c[2*p+1]*LOG2E);
}

// ---------------------------------------------------------------------------
// Stage 2: P x G contraction.  Each wave owns 32 points (two 16-row WMMA
// tiles sharing B operands) and sweeps 2048 gaussians in 128 column tiles.
// Per tile: 3 float2 loads, 6 v_wmma_f32_16x16x4_f32, 16 v_exp_f32, 16 adds.
// ---------------------------------------------------------------------------
__global__ __launch_bounds__(256) void gs_eval(
    const float* __restrict__ x, const float2* __restrict__ cpack,
    const unsigned char* __restrict__ inside, float* __restrict__ out)
{
    const int lane  = threadIdx.x & 31;
    const int wave  = threadIdx.x >> 5;
    const int half  = lane >> 4;     // 0: lanes 0-15, 1: lanes 16-31
    const int l16   = lane & 15;
    const int pbase = blockIdx.x * 256 + wave * 32;   // 8 waves * 32 pts

    // Build loop-invariant A operands per the ISA 16x4 f32 layout:
    //   VGPR0: K = 4s+0 (lanes 0-15) / 4s+2 (lanes 16-31); VGPR1: +1
    v2f av[2][3];
#pragma unroll
    for (int r = 0; r < 2; ++r) {
        int p = pbase + r*16 + l16;
        float px = (x[p*3+0] + 1.0f) * 0.5f;
        float py = (x[p*3+1] + 1.0f) * 0.5f;
        float pz = (x[p*3+2] + 1.0f) * 0.5f;
        float f[12] = { px*px, py*py, pz*pz, px*py, px*pz, py*pz,
                        px, py, pz, 1.0f, 0.0f, 0.0f };
#pragma unroll
        for (int s = 0; s < 3; ++s) {
            av[r][s].x = half ? f[4*s+2] : f[4*s+0];   // v_cndmask, no scratch
            av[r][s].y = half ? f[4*s+3] : f[4*s+1];
        }
    }

    v8f acc0 = {};   // running exp-sums, tile 0 (rows pbase..pbase+15)
    v8f acc1 = {};   // tile 1 (rows pbase+16..pbase+31)

    // B layout mirrors A: VGPR0 holds K=4s+2*half, VGPR1 holds K=4s+2*half+1
    // -> one float2 per slice per lane; 16 lanes read 128B contiguous.
    const float2* __restrict__ bptr = cpack + half*G_N + l16;

    for (int t = 0; t < G_N/16; ++t) {
        v2f b[3];
#pragma unroll
        for (int s = 0; s < 3; ++s) {
            float2 ld = bptr[s*2*G_N + t*16];
            b[s].x = ld.x; b[s].y = ld.y;
        }
        v8f p0 = {};
        v8f p1 = {};
#pragma unroll
        for (int s = 0; s < 3; ++s) {
            p0 = __builtin_amdgcn_wmma_f32_16x16x4_f32(
                    false, av[0][s], false, b[s], (short)0, p0, false, false);
            p1 = __builtin_amdgcn_wmma_f32_16x16x4_f32(
                    false, av[1][s], false, b[s], (short)0, p1, false, false);
        }
        // epilogue: hardware exp2 (coefficients pre-scaled by log2 e)
#pragma unroll
        for (int i = 0; i < 8; ++i) {
            acc0[i] += __builtin_amdgcn_exp2f(p0[i]);
            acc1[i] += __builtin_amdgcn_exp2f(p1[i]);
        }
    }

    // Column reduction: C/D layout -> lanes 0-15 hold rows 0-7 (VGPR i = row i),
    // lanes 16-31 hold rows 8-15. XOR-reduce within each 16-lane half.
#pragma unroll
    for (int i = 0; i < 8; ++i) {
        float a0 = acc0[i], a1 = acc1[i];
#pragma unroll
        for (int m = 1; m < 16; m <<= 1) {
            a0 += __shfl_xor(a0, m, 32);
            a1 += __shfl_xor(a1, m, 32);
        }
        acc0[i] = a0; acc1[i] = a1;
    }

    if (l16 == 0) {           // lane 0 writes rows 0-7, lane 16 rows 8-15
        int rowbase = pbase + half*8;
#pragma unroll
        for (int i = 0; i < 8; ++i) {
            int pa = rowbase + i;
            out[pa] = inside[pa] ? acc0[i] : 0.0f;
            int pb = rowbase + 16 + i;
            out[pb] = inside[pb] ? acc1[i] : 0.0f;
        }
    }
}

// ---------------------------------------------------------------------------
extern "C" void kernel_launch(void* const* d_in, const int* in_sizes, int n_in,
                              void* d_out, int out_size, void* d_ws, size_t ws_size,
                              hipStream_t stream)
{
    const float* x        = (const float*)d_in[0];   // (P,3)
    const float* xyz      = (const float*)d_in[1];   // (G,3)
    const float* weight   = (const float*)d_in[2];   // (G,1)
    const float* scaling  = (const float*)d_in[3];   // (G,3)
    const float* rotation = (const float*)d_in[4];   // (G,4)
    const float* values   = (const float*)d_in[5];   // (G,1)
    const unsigned char* inside = (const unsigned char*)d_in[6]; // (P,) bool
    float* out = (float*)d_out;                      // (P,1)

    float2* cpack = (float2*)d_ws;                   // 6*G float2 = 96 KB

    gs_precompute<<<G_N/256, 256, 0, stream>>>(xyz, weight, scaling, rotation,
                                               values, cpack);
    gs_eval<<<P_N/256, 256, 0, stream>>>(x, cpack, inside, out);
}